// Edge2NodeLayer_15693810499830
// MI455X (gfx1250) — compile-verified
//
#include <hip/hip_runtime.h>
#include <hip/hip_bf16.h>

// Problem constants (from reference): B=64, F=240 -> 15360 frames; J=25; CIN=32; COUT=64.
#define NFRAMES 15360
#define JN 25
#define CIN 32
#define COUT 64
#define WAVES_PER_BLOCK 4

typedef __attribute__((ext_vector_type(2))) float v2f;
typedef __attribute__((ext_vector_type(8))) float v8f;

static __device__ __forceinline__ v8f wmma_f32(v2f a, v2f b, v8f c) {
    // V_WMMA_F32_16X16X4_F32: D = A(16x4 f32) x B(4x16 f32) + C(16x16 f32)
    return __builtin_amdgcn_wmma_f32_16x16x4_f32(
        /*neg_a=*/false, a, /*neg_b=*/false, b,
        /*c_mod=*/(short)0, c, /*reuse_a=*/false, /*reuse_b=*/false);
}

// A-fragment from a row-major [nrows x 32] fp32 matrix (K fully in-bounds, rows masked).
// Lane layout (ISA 7.12.2, 32-bit A 16x4): lanes 0-15: M=lane, K={k0,k0+1};
// lanes 16-31: M=lane-16, K={k0+2,k0+3}. K pairs are contiguous & 8B aligned.
static __device__ __forceinline__ v2f loadA_rows32(const float* __restrict__ base,
                                                   int mtile, int k0, int lane) {
    const int half = lane >> 4;
    const int m = mtile * 16 + (lane & 15);
    const int k = k0 + 2 * half;
    v2f a;
    if (m < JN) {
        const float* p = base + m * CIN + k;
        a.x = p[0];
        a.y = p[1];
    } else {
        a.x = 0.0f; a.y = 0.0f;
    }
    return a;
}

// A-fragment from adjacency [25 x 25] (both rows and cols masked; stride 25 => scalar loads).
static __device__ __forceinline__ v2f loadA_adj(const float* __restrict__ aF,
                                                int mtile, int k0, int lane) {
    const int half = lane >> 4;
    const int m = mtile * 16 + (lane & 15);
    const int k = k0 + 2 * half;
    v2f a;
    const bool mv = (m < JN);
    a.x = (mv && (k     < JN)) ? aF[m * JN + k]     : 0.0f;
    a.y = (mv && (k + 1 < JN)) ? aF[m * JN + k + 1] : 0.0f;
    return a;
}

// B-fragment from a row-major [K x 64] fp32 matrix (weights in global, or EC tile in LDS).
// Lanes 0-15: N=lane, K={k0,k0+1}; lanes 16-31: N=lane-16, K={k0+2,k0+3}.
template <typename PTR>
static __device__ __forceinline__ v2f loadB_64(PTR W, int k0, int ntile, int lane) {
    const int half = lane >> 4;
    const int n = ntile * 16 + (lane & 15);
    const int k = k0 + 2 * half;
    v2f b;
    b.x = W[k * COUT + n];
    b.y = W[(k + 1) * COUT + n];
    return b;
}

__global__ __launch_bounds__(WAVES_PER_BLOCK * 32)
void edge2node_fused_kernel(const float* __restrict__ node,
                            const float* __restrict__ edge,
                            const float* __restrict__ adj,
                            const float* __restrict__ Wp, const float* __restrict__ bp,
                            const float* __restrict__ Wr, const float* __restrict__ br,
                            const float* __restrict__ Wc, const float* __restrict__ bc,
                            float* __restrict__ out) {
    // Per-wave LDS: EC tile (32x64 fp32, rows 25..31 hold bc -> annihilated by zero adj cols)
    // and num_neighbours (25 used).
    __shared__ float ecBuf[WAVES_PER_BLOCK][32 * COUT];
    __shared__ float nnBuf[WAVES_PER_BLOCK][32];

    const int wave  = (int)(threadIdx.x >> 5);
    const int lane  = (int)(threadIdx.x & 31);
    const int frame = (int)blockIdx.x * WAVES_PER_BLOCK + wave;
    if (frame >= NFRAMES) return;  // wave-uniform

    const float* eF = edge + (size_t)frame * JN * CIN;
    const float* nF = node + (size_t)frame * JN * CIN;
    const float* aF = adj  + (size_t)frame * JN * JN;
    float*       oF = out  + (size_t)frame * JN * COUT;

    float* ecw = &ecBuf[wave][0];

    // ---- degrees: num_neighbours[m] = rowsum(adj[m,:]) + colsum(adj[:,m]) + 1 ----
    if (lane < JN) {
        float rs = 0.0f, cs = 0.0f;
        for (int j = 0; j < JN; ++j) {
            rs += aF[lane * JN + j];
            cs += aF[j * JN + lane];
        }
        nnBuf[wave][lane] = rs + cs + 1.0f;
    }

    const int half = lane >> 4;
    const int nr   = lane & 15;

    // ---- pass 1: EC = E @ Wc + bc, staged into LDS (row-major 32x64) ----
    for (int mtile = 0; mtile < 2; ++mtile) {
        for (int ntile = 0; ntile < 4; ++ntile) {
            v8f acc = {};
#pragma unroll
            for (int k0 = 0; k0 < CIN; k0 += 4) {
                v2f a = loadA_rows32(eF, mtile, k0, lane);
                v2f b = loadB_64<const float*>(Wc, k0, ntile, lane);
                acc = wmma_f32(a, b, acc);
            }
            const int n = ntile * 16 + nr;
            const float bcn = bc[n];
#pragma unroll
            for (int r = 0; r < 8; ++r) {
                const int m = mtile * 16 + r + 8 * half;  // C/D layout
                ecw[m * COUT + n] = acc[r] + bcn;
            }
        }
    }
    // Same-wave LDS RAW: DS ops are in-order per wave; compiler inserts s_wait_dscnt.

    // ---- pass 2: parent/recurrent GEMMs + adj @ EC, combine, divide, store ----
    for (int mtile = 0; mtile < 2; ++mtile) {
        for (int ntile = 0; ntile < 4; ++ntile) {
            v8f accP = {};  // E @ Wp
            v8f accR = {};  // N @ Wr
            v8f accC = {};  // adj @ EC
#pragma unroll
            for (int k0 = 0; k0 < CIN; k0 += 4) {
                v2f ae = loadA_rows32(eF, mtile, k0, lane);
                v2f bp2 = loadB_64<const float*>(Wp, k0, ntile, lane);
                accP = wmma_f32(ae, bp2, accP);
                v2f an = loadA_rows32(nF, mtile, k0, lane);
                v2f br2 = loadB_64<const float*>(Wr, k0, ntile, lane);
                accR = wmma_f32(an, br2, accR);
            }
#pragma unroll
            for (int k0 = 0; k0 < 28; k0 += 4) {  // adj K = 25, padded cols are zero
                v2f aa = loadA_adj(aF, mtile, k0, lane);
                v2f bb = loadB_64<const float*>(ecw, k0, ntile, lane);
                accC = wmma_f32(aa, bb, accC);
            }
            const int n = ntile * 16 + nr;
            const float bpn = bp[n];
            const float brn = br[n];
#pragma unroll
            for (int r = 0; r < 8; ++r) {
                const int m = mtile * 16 + r + 8 * half;
                if (m < JN) {
                    const float val = (5.0f * (accP[r] + bpn) + 5.0f * accC[r]
                                       + accR[r] + brn) / nnBuf[wave][m];
                    oF[m * COUT + n] = val;
                }
            }
        }
    }
}

extern "C" void kernel_launch(void* const* d_in, const int* in_sizes, int n_in,
                              void* d_out, int out_size, void* d_ws, size_t ws_size,
                              hipStream_t stream) {
    // setup_inputs order: node_features, edge_features, adj_matrix, Wp, bp, Wr, br, Wc, bc
    const float* node = (const float*)d_in[0];
    const float* edge = (const float*)d_in[1];
    const float* adjm = (const float*)d_in[2];
    const float* Wp   = (const float*)d_in[3];
    const float* bp   = (const float*)d_in[4];
    const float* Wr   = (const float*)d_in[5];
    const float* br   = (const float*)d_in[6];
    const float* Wc   = (const float*)d_in[7];
    const float* bc   = (const float*)d_in[8];
    float* out = (float*)d_out;

    dim3 block(WAVES_PER_BLOCK * 32);
    dim3 grid((NFRAMES + WAVES_PER_BLOCK - 1) / WAVES_PER_BLOCK);  // 3840
    hipLaunchKernelGGL(edge2node_fused_kernel, grid, block, 0, stream,
                       node, edge, adjm, Wp, bp, Wr, br, Wc, bc, out);
}